// GIN_6640019439791
// MI455X (gfx1250) — compile-verified
//
#include <hip/hip_runtime.h>
#include <hip/hip_bf16.h>

typedef __attribute__((ext_vector_type(2))) float v2f;
typedef __attribute__((ext_vector_type(8))) float v8f;

#define NN 100000      // nodes
#define NE 1600000     // edges
#define DF 64          // feature dim
#define NL 3           // GNN layers
#define NG 128         // graphs
#define NC 10          // classes
#define BN_EPS 1e-5f

// ---------------------------------------------------------------------------
// zero a float buffer
// ---------------------------------------------------------------------------
__global__ void zero_f32_kernel(float* __restrict__ p, int n) {
  int i = blockIdx.x * blockDim.x + threadIdx.x;
  if (i < n) p[i] = 0.0f;
}

// ---------------------------------------------------------------------------
// agg[dst[e]] += h[src[e]]  — 16 threads per edge, float4 per thread.
// Random 256B row gathers + f32 atomics; agg (25.6MB) lives in L2 (192MB).
// ---------------------------------------------------------------------------
__global__ void edge_aggregate_kernel(const float* __restrict__ h,
                                      const int* __restrict__ src,
                                      const int* __restrict__ dst,
                                      float* __restrict__ agg) {
  int gid = blockIdx.x * blockDim.x + threadIdx.x;
  if (gid >= NE * 16) return;
  int e = gid >> 4;
  int f = (gid & 15) * 4;
  int s = src[e];
  int d = dst[e];
  const float4 v = *(const float4*)(h + (long long)s * DF + f);
  float* a = agg + (long long)d * DF + f;
  atomicAdd(a + 0, v.x);
  atomicAdd(a + 1, v.y);
  atomicAdd(a + 2, v.z);
  atomicAdd(a + 3, v.w);
}

// ---------------------------------------------------------------------------
// One GIN layer, fully fused:
//   z = relu((h+agg) @ W1 + b1); z = z*bn_scale + bn_shift; h_out = z @ W2 + b2
// 8 waves/block; each wave computes a 16-row tile across all 64 columns using
// V_WMMA_F32_16X16X4_F32 (fp32 in/out -> exact fp32 semantics).
// ---------------------------------------------------------------------------
__global__ __launch_bounds__(256, 1) void gin_layer_kernel(
    const float* __restrict__ h_in, const float* __restrict__ agg,
    const float* __restrict__ W1, const float* __restrict__ b1,
    const float* __restrict__ gamma, const float* __restrict__ beta,
    const float* __restrict__ rm, const float* __restrict__ rv,
    const float* __restrict__ W2, const float* __restrict__ b2,
    float* __restrict__ h_out) {
  __shared__ float sW1[DF * DF];        // 16 KB
  __shared__ float sW2[DF * DF];        // 16 KB
  __shared__ float sZ[8 * 16 * DF];     // 32 KB: per-wave z tiles

  const int tid = threadIdx.x;
  for (int i = tid; i < DF * DF; i += 256) {
    sW1[i] = W1[i];
    sW2[i] = W2[i];
  }
  __syncthreads();

  const int wave  = tid >> 5;
  const int lane  = tid & 31;
  const int lrow  = lane & 15;            // A row / C column within tile
  const int khalf = (lane >> 4) << 1;     // 0 or 2 (K sub-offset per lane half)
  const int mhalf = (lane >> 4) << 3;     // 0 or 8 (C row base per lane half)
  const int tile  = blockIdx.x * 128 + wave * 16;
  const int rowc  = min(tile + lrow, NN - 1);   // clamp loads; EXEC stays full
  const float* xh = h_in + (long long)rowc * DF;
  const float* xa = agg + (long long)rowc * DF;

  // epilogue params for the 4 N-tiles (column n = nt*16 + lrow)
  float bias1[4], bias2[4], bns[4], bnb[4];
#pragma unroll
  for (int nt = 0; nt < 4; ++nt) {
    int n = nt * 16 + lrow;
    float s = gamma[n] * rsqrtf(rv[n] + BN_EPS);
    bns[nt] = s;
    bnb[nt] = beta[n] - rm[n] * s;
    bias1[nt] = b1[n];
    bias2[nt] = b2[n];
  }

  // ---- GEMM1: (h+agg) @ W1 ----
  v8f acc[4] = {};
#pragma unroll
  for (int k0 = 0; k0 < DF; k0 += 4) {
    int kk = k0 + khalf;
    v2f a;
    a.x = xh[kk] + xa[kk];
    a.y = xh[kk + 1] + xa[kk + 1];
    const float* w0 = sW1 + kk * DF + lrow;
    const float* w1 = sW1 + (kk + 1) * DF + lrow;
#pragma unroll
    for (int nt = 0; nt < 4; ++nt) {
      v2f b;
      b.x = w0[nt * 16];
      b.y = w1[nt * 16];
      acc[nt] = __builtin_amdgcn_wmma_f32_16x16x4_f32(
          false, a, false, b, (short)0, acc[nt], false, false);
    }
  }

  // relu + batchnorm(eval), stage z tile to LDS (C-layout -> row-major)
  float* zt = sZ + wave * 16 * DF;
#pragma unroll
  for (int v = 0; v < 8; ++v) {
    int m = mhalf + v;
#pragma unroll
    for (int nt = 0; nt < 4; ++nt) {
      float z = acc[nt][v] + bias1[nt];
      z = fmaxf(z, 0.0f);
      z = z * bns[nt] + bnb[nt];
      zt[m * DF + nt * 16 + lrow] = z;
    }
  }
  __syncthreads();

  // ---- GEMM2: z @ W2 ----
  v8f acc2[4] = {};
#pragma unroll
  for (int k0 = 0; k0 < DF; k0 += 4) {
    int kk = k0 + khalf;
    v2f a;
    a.x = zt[lrow * DF + kk];
    a.y = zt[lrow * DF + kk + 1];
    const float* w0 = sW2 + kk * DF + lrow;
    const float* w1 = sW2 + (kk + 1) * DF + lrow;
#pragma unroll
    for (int nt = 0; nt < 4; ++nt) {
      v2f b;
      b.x = w0[nt * 16];
      b.y = w1[nt * 16];
      acc2[nt] = __builtin_amdgcn_wmma_f32_16x16x4_f32(
          false, a, false, b, (short)0, acc2[nt], false, false);
    }
  }

#pragma unroll
  for (int v = 0; v < 8; ++v) {
    int r = tile + mhalf + v;
    if (r < NN) {
#pragma unroll
      for (int nt = 0; nt < 4; ++nt) {
        h_out[(long long)r * DF + nt * 16 + lrow] = acc2[nt][v] + bias2[nt];
      }
    }
  }
}

// ---------------------------------------------------------------------------
// global_add_pool via atomics: pooled[batch[n]] += h[n]
// ---------------------------------------------------------------------------
__global__ void pool_kernel(const float* __restrict__ h,
                            const int* __restrict__ batch,
                            float* __restrict__ pooled) {
  int gid = blockIdx.x * blockDim.x + threadIdx.x;
  if (gid >= NN * 16) return;
  int n = gid >> 4;
  int f = (gid & 15) * 4;
  int g = batch[n];
  const float4 v = *(const float4*)(h + (long long)n * DF + f);
  float* p = pooled + g * DF + f;
  atomicAdd(p + 0, v.x);
  atomicAdd(p + 1, v.y);
  atomicAdd(p + 2, v.z);
  atomicAdd(p + 3, v.w);
}

// ---------------------------------------------------------------------------
// classifier head: relu(pooled @ Wc1 + bc1) @ Wc2 + bc2 -> log_softmax
// One block, 8 waves; GEMM1 on WMMA, tiny 64x10 GEMM + softmax scalar-side.
// ---------------------------------------------------------------------------
__global__ __launch_bounds__(256, 1) void classifier_kernel(
    const float* __restrict__ pooled,
    const float* __restrict__ Wc1, const float* __restrict__ bc1,
    const float* __restrict__ Wc2, const float* __restrict__ bc2,
    float* __restrict__ out) {
  __shared__ float sW[DF * DF];   // 16 KB
  __shared__ float sZ[NG * DF];   // 32 KB
  const int tid = threadIdx.x;
  for (int i = tid; i < DF * DF; i += 256) sW[i] = Wc1[i];
  __syncthreads();

  const int wave  = tid >> 5;
  const int lane  = tid & 31;
  const int lrow  = lane & 15;
  const int khalf = (lane >> 4) << 1;
  const int mhalf = (lane >> 4) << 3;
  const int rbase = wave * 16;    // 8 waves x 16 rows = 128 graphs

  v8f acc[4] = {};
#pragma unroll
  for (int k0 = 0; k0 < DF; k0 += 4) {
    int kk = k0 + khalf;
    v2f a;
    a.x = pooled[(rbase + lrow) * DF + kk];
    a.y = pooled[(rbase + lrow) * DF + kk + 1];
    const float* w0 = sW + kk * DF + lrow;
    const float* w1 = sW + (kk + 1) * DF + lrow;
#pragma unroll
    for (int nt = 0; nt < 4; ++nt) {
      v2f b;
      b.x = w0[nt * 16];
      b.y = w1[nt * 16];
      acc[nt] = __builtin_amdgcn_wmma_f32_16x16x4_f32(
          false, a, false, b, (short)0, acc[nt], false, false);
    }
  }
#pragma unroll
  for (int v = 0; v < 8; ++v) {
    int m = rbase + mhalf + v;
#pragma unroll
    for (int nt = 0; nt < 4; ++nt) {
      int n = nt * 16 + lrow;
      float z = acc[nt][v] + bc1[n];
      sZ[m * DF + n] = fmaxf(z, 0.0f);
    }
  }
  __syncthreads();

  if (tid < NG) {
    float logit[NC];
    float mx = -3.402823466e38f;
    for (int c = 0; c < NC; ++c) {
      float s = bc2[c];
      for (int k = 0; k < DF; ++k) s += sZ[tid * DF + k] * Wc2[k * NC + c];
      logit[c] = s;
      mx = fmaxf(mx, s);
    }
    float sum = 0.0f;
    for (int c = 0; c < NC; ++c) sum += expf(logit[c] - mx);
    float lse = mx + logf(sum);
    for (int c = 0; c < NC; ++c) out[tid * NC + c] = logit[c] - lse;
  }
}

// ---------------------------------------------------------------------------
extern "C" void kernel_launch(void* const* d_in, const int* in_sizes, int n_in,
                              void* d_out, int out_size, void* d_ws,
                              size_t ws_size, hipStream_t stream) {
  (void)in_sizes; (void)n_in; (void)out_size; (void)ws_size;
  const float* x     = (const float*)d_in[0];
  const float* W1    = (const float*)d_in[1];
  const float* b1    = (const float*)d_in[2];
  const float* gamma = (const float*)d_in[3];
  const float* beta  = (const float*)d_in[4];
  const float* rm    = (const float*)d_in[5];
  const float* rv    = (const float*)d_in[6];
  const float* W2    = (const float*)d_in[7];
  const float* b2    = (const float*)d_in[8];
  const float* Wc1   = (const float*)d_in[9];
  const float* bc1   = (const float*)d_in[10];
  const float* Wc2   = (const float*)d_in[11];
  const float* bc2   = (const float*)d_in[12];
  const int* ei      = (const int*)d_in[13];
  const int* batch   = (const int*)d_in[14];
  const int* src = ei;            // edge_index[0]
  const int* dst = ei + NE;       // edge_index[1]

  // workspace: hA | hB | agg | pooled  (~77 MB of f32)
  float* ws = (float*)d_ws;
  float* hA = ws;
  float* hB = ws + (size_t)NN * DF;
  float* agg = ws + 2 * (size_t)NN * DF;
  float* pooled = ws + 3 * (size_t)NN * DF;

  const float* hin = x;
  float* hout = hA;
  for (int l = 0; l < NL; ++l) {
    zero_f32_kernel<<<(NN * DF + 255) / 256, 256, 0, stream>>>(agg, NN * DF);
    edge_aggregate_kernel<<<(NE * 16 + 255) / 256, 256, 0, stream>>>(
        hin, src, dst, agg);
    gin_layer_kernel<<<(NN + 127) / 128, 256, 0, stream>>>(
        hin, agg, W1 + l * DF * DF, b1 + l * DF, gamma + l * DF, beta + l * DF,
        rm + l * DF, rv + l * DF, W2 + l * DF * DF, b2 + l * DF, hout);
    hin = hout;
    hout = (hout == hA) ? hB : hA;
  }

  zero_f32_kernel<<<(NG * DF + 255) / 256, 256, 0, stream>>>(pooled, NG * DF);
  pool_kernel<<<(NN * 16 + 255) / 256, 256, 0, stream>>>(hin, batch, pooled);
  classifier_kernel<<<1, 256, 0, stream>>>(pooled, Wc1, bc1, Wc2, bc2,
                                           (float*)d_out);
}